// L0MLLP_1520418423392
// MI455X (gfx1250) — compile-verified
//
#include <hip/hip_runtime.h>

// Problem dims (fixed by reference)
#define BB   256
#define IN   512
#define HID  1024
#define OUT  512

#define LIMIT_A -0.1f
#define LIMIT_B  1.1f

typedef __attribute__((ext_vector_type(2))) float v2f;
typedef __attribute__((ext_vector_type(8))) float v8f;

// Hard-concrete deterministic gate: clamp(sigmoid(q)*(b-a)+a, 0, 1)
__device__ __forceinline__ float hc_gate(float q) {
    float pi = 1.0f / (1.0f + __expf(-q));
    float z  = pi * (LIMIT_B - LIMIT_A) + LIMIT_A;
    return fminf(fmaxf(z, 0.0f), 1.0f);
}

// xin1[b,i] = x[b,i] * gate(qz1[i])
__global__ void __launch_bounds__(256)
l0_gate_xin(const float* __restrict__ x, const float* __restrict__ qz1,
            float* __restrict__ xin, int n) {
    int idx = blockIdx.x * blockDim.x + threadIdx.x;
    if (idx < n) {
        int i = idx & (IN - 1);
        xin[idx] = x[idx] * hc_gate(qz1[i]);
    }
}

// z[i] = gate(qz[i])
__global__ void __launch_bounds__(256)
l0_gate(const float* __restrict__ qz, float* __restrict__ z, int n) {
    int idx = blockIdx.x * blockDim.x + threadIdx.x;
    if (idx < n) z[idx] = hc_gate(qz[idx]);
}

// -----------------------------------------------------------------------------
// Log-domain series GEMM on f32 WMMA, 32x16 output tile per wave (two 16x16
// accumulators sharing the strided B loads + B power chain).
//   acc[b,j] = sum_{k=1..6} sum_i a[b,i]^k * ( W[i,j]^k / k )
// Layer 1 epilogue: h = exp(-acc).   Layer 2 epilogue: out = 1 - exp(-acc).
// -----------------------------------------------------------------------------

__global__ void __launch_bounds__(128)
l0_conj_wmma(const float* __restrict__ A, const float* __restrict__ W,
             float* __restrict__ H) {
    const int lane = threadIdx.x & 31;
    const int wave = threadIdx.x >> 5;
    const int tile = blockIdx.x * 4 + wave;          // 512 tiles (8 x 64)
    const int ntn  = HID / 16;
    const int tm   = tile / ntn;                     // 32-row band
    const int tn   = tile - tm * ntn;

    const int mn = lane & 15;          // A-row (lanes) / B-col (lanes)
    const int kp = (lane >> 4) << 1;   // lanes 0-15 -> K 0,1 ; lanes 16-31 -> K 2,3

    const float* arow0 = A + (tm * 32 + mn) * IN + kp;
    const float* arow1 = arow0 + 16 * IN;
    const float* bcol  = W + tn * 16 + mn;

    v8f acc0 = {0.f, 0.f, 0.f, 0.f, 0.f, 0.f, 0.f, 0.f};
    v8f acc1 = {0.f, 0.f, 0.f, 0.f, 0.f, 0.f, 0.f, 0.f};

    for (int i0 = 0; i0 < IN; i0 += 4) {
        __builtin_prefetch(arow0 + i0 + 64, 0, 0);  // global_prefetch_b8
        v2f a0 = *(const v2f*)(arow0 + i0);
        v2f a1 = *(const v2f*)(arow1 + i0);
        v2f b;
        b.x = bcol[(i0 + kp) * HID];
        b.y = bcol[(i0 + kp + 1) * HID];

        // B powers with 1/k folded in (computed once, shared by both tiles)
        v2f b2 = b * b,   bs2 = b2 * 0.5f;
        v2f b3 = b2 * b,  bs3 = b3 * (1.0f / 3.0f);
        v2f b4 = b2 * b2, bs4 = b4 * 0.25f;
        v2f b5 = b4 * b,  bs5 = b5 * 0.2f;
        v2f b6 = b3 * b3, bs6 = b6 * (1.0f / 6.0f);

        // A powers, per row-tile (plain powers; scale lives on B)
        v2f a0_2 = a0 * a0,     a1_2 = a1 * a1;
        v2f a0_3 = a0_2 * a0,   a1_3 = a1_2 * a1;
        v2f a0_4 = a0_2 * a0_2, a1_4 = a1_2 * a1_2;
        v2f a0_5 = a0_4 * a0,   a1_5 = a1_4 * a1;
        v2f a0_6 = a0_3 * a0_3, a1_6 = a1_3 * a1_3;

        acc0 = __builtin_amdgcn_wmma_f32_16x16x4_f32(false, a0,   false, b,   (short)0, acc0, false, false);
        acc1 = __builtin_amdgcn_wmma_f32_16x16x4_f32(false, a1,   false, b,   (short)0, acc1, false, false);
        acc0 = __builtin_amdgcn_wmma_f32_16x16x4_f32(false, a0_2, false, bs2, (short)0, acc0, false, false);
        acc1 = __builtin_amdgcn_wmma_f32_16x16x4_f32(false, a1_2, false, bs2, (short)0, acc1, false, false);
        acc0 = __builtin_amdgcn_wmma_f32_16x16x4_f32(false, a0_3, false, bs3, (short)0, acc0, false, false);
        acc1 = __builtin_amdgcn_wmma_f32_16x16x4_f32(false, a1_3, false, bs3, (short)0, acc1, false, false);
        acc0 = __builtin_amdgcn_wmma_f32_16x16x4_f32(false, a0_4, false, bs4, (short)0, acc0, false, false);
        acc1 = __builtin_amdgcn_wmma_f32_16x16x4_f32(false, a1_4, false, bs4, (short)0, acc1, false, false);
        acc0 = __builtin_amdgcn_wmma_f32_16x16x4_f32(false, a0_5, false, bs5, (short)0, acc0, false, false);
        acc1 = __builtin_amdgcn_wmma_f32_16x16x4_f32(false, a1_5, false, bs5, (short)0, acc1, false, false);
        acc0 = __builtin_amdgcn_wmma_f32_16x16x4_f32(false, a0_6, false, bs6, (short)0, acc0, false, false);
        acc1 = __builtin_amdgcn_wmma_f32_16x16x4_f32(false, a1_6, false, bs6, (short)0, acc1, false, false);
    }

    // C/D layout: VGPR r -> row (r + 8*(lane>=16)), col = lane&15
    const int mbase = tm * 32 + ((lane >> 4) << 3);
    const int nidx  = tn * 16 + mn;
#pragma unroll
    for (int r = 0; r < 8; ++r) {
        H[(mbase + r) * HID + nidx]        = __expf(-acc0[r]);
        H[(mbase + 16 + r) * HID + nidx]   = __expf(-acc1[r]);
    }
}

__global__ void __launch_bounds__(128)
l0_disj_wmma(const float* __restrict__ H, const float* __restrict__ z2,
             const float* __restrict__ W, float* __restrict__ Oout) {
    const int lane = threadIdx.x & 31;
    const int wave = threadIdx.x >> 5;
    const int tile = blockIdx.x * 4 + wave;          // 256 tiles (8 x 32)
    const int ntn  = OUT / 16;
    const int tm   = tile / ntn;
    const int tn   = tile - tm * ntn;

    const int mn = lane & 15;
    const int kp = (lane >> 4) << 1;

    const float* arow0 = H + (tm * 32 + mn) * HID + kp;
    const float* arow1 = arow0 + 16 * HID;
    const float* bcol  = W + tn * 16 + mn;
    const float* zrow  = z2 + kp;

    v8f acc0 = {0.f, 0.f, 0.f, 0.f, 0.f, 0.f, 0.f, 0.f};
    v8f acc1 = {0.f, 0.f, 0.f, 0.f, 0.f, 0.f, 0.f, 0.f};

    for (int i0 = 0; i0 < HID; i0 += 4) {
        __builtin_prefetch(arow0 + i0 + 64, 0, 0);
        v2f z  = *(const v2f*)(zrow + i0);           // gate, shared
        v2f a0 = *(const v2f*)(arow0 + i0) * z;
        v2f a1 = *(const v2f*)(arow1 + i0) * z;
        v2f b;
        b.x = bcol[(i0 + kp) * OUT];
        b.y = bcol[(i0 + kp + 1) * OUT];

        v2f b2 = b * b,   bs2 = b2 * 0.5f;
        v2f b3 = b2 * b,  bs3 = b3 * (1.0f / 3.0f);
        v2f b4 = b2 * b2, bs4 = b4 * 0.25f;
        v2f b5 = b4 * b,  bs5 = b5 * 0.2f;
        v2f b6 = b3 * b3, bs6 = b6 * (1.0f / 6.0f);

        v2f a0_2 = a0 * a0,     a1_2 = a1 * a1;
        v2f a0_3 = a0_2 * a0,   a1_3 = a1_2 * a1;
        v2f a0_4 = a0_2 * a0_2, a1_4 = a1_2 * a1_2;
        v2f a0_5 = a0_4 * a0,   a1_5 = a1_4 * a1;
        v2f a0_6 = a0_3 * a0_3, a1_6 = a1_3 * a1_3;

        acc0 = __builtin_amdgcn_wmma_f32_16x16x4_f32(false, a0,   false, b,   (short)0, acc0, false, false);
        acc1 = __builtin_amdgcn_wmma_f32_16x16x4_f32(false, a1,   false, b,   (short)0, acc1, false, false);
        acc0 = __builtin_amdgcn_wmma_f32_16x16x4_f32(false, a0_2, false, bs2, (short)0, acc0, false, false);
        acc1 = __builtin_amdgcn_wmma_f32_16x16x4_f32(false, a1_2, false, bs2, (short)0, acc1, false, false);
        acc0 = __builtin_amdgcn_wmma_f32_16x16x4_f32(false, a0_3, false, bs3, (short)0, acc0, false, false);
        acc1 = __builtin_amdgcn_wmma_f32_16x16x4_f32(false, a1_3, false, bs3, (short)0, acc1, false, false);
        acc0 = __builtin_amdgcn_wmma_f32_16x16x4_f32(false, a0_4, false, bs4, (short)0, acc0, false, false);
        acc1 = __builtin_amdgcn_wmma_f32_16x16x4_f32(false, a1_4, false, bs4, (short)0, acc1, false, false);
        acc0 = __builtin_amdgcn_wmma_f32_16x16x4_f32(false, a0_5, false, bs5, (short)0, acc0, false, false);
        acc1 = __builtin_amdgcn_wmma_f32_16x16x4_f32(false, a1_5, false, bs5, (short)0, acc1, false, false);
        acc0 = __builtin_amdgcn_wmma_f32_16x16x4_f32(false, a0_6, false, bs6, (short)0, acc0, false, false);
        acc1 = __builtin_amdgcn_wmma_f32_16x16x4_f32(false, a1_6, false, bs6, (short)0, acc1, false, false);
    }

    const int mbase = tm * 32 + ((lane >> 4) << 3);
    const int nidx  = tn * 16 + mn;
#pragma unroll
    for (int r = 0; r < 8; ++r) {
        Oout[(mbase + r) * OUT + nidx]      = 1.0f - __expf(-acc0[r]);
        Oout[(mbase + 16 + r) * OUT + nidx] = 1.0f - __expf(-acc1[r]);
    }
}

extern "C" void kernel_launch(void* const* d_in, const int* in_sizes, int n_in,
                              void* d_out, int out_size, void* d_ws, size_t ws_size,
                              hipStream_t stream) {
    const float* x   = (const float*)d_in[0];   // [B, IN]
    const float* W1  = (const float*)d_in[1];   // [IN, HID]
    const float* qz1 = (const float*)d_in[2];   // [IN]
    const float* W2  = (const float*)d_in[3];   // [HID, OUT]
    const float* qz2 = (const float*)d_in[4];   // [HID]
    float* out = (float*)d_out;                 // [B, OUT]

    // Workspace layout (floats): xin1 [B*IN] | h [B*HID] | z2 [HID]
    float* xin1 = (float*)d_ws;
    float* h    = xin1 + (size_t)BB * IN;
    float* z2   = h + (size_t)BB * HID;

    // 1. gates folded into x
    l0_gate_xin<<<(BB * IN) / 256, 256, 0, stream>>>(x, qz1, xin1, BB * IN);
    // 2. layer-2 gate vector
    l0_gate<<<HID / 256, 256, 0, stream>>>(qz2, z2, HID);
    // 3. conjunction layer: 8x64 32-row tiles, 4 waves/block
    l0_conj_wmma<<<((BB / 32) * (HID / 16)) / 4, 128, 0, stream>>>(xin1, W1, h);
    // 4. disjunction layer: 8x32 32-row tiles, 4 waves/block
    l0_disj_wmma<<<((BB / 32) * (OUT / 16)) / 4, 128, 0, stream>>>(h, z2, W2, out);
}